// IOULoss_687194767538
// MI455X (gfx1250) — compile-verified
//
#include <hip/hip_runtime.h>
#include <hip/hip_bf16.h>
#include <stdint.h>

#define MAX_T 512

typedef float v8f  __attribute__((ext_vector_type(8)));
typedef float v2f  __attribute__((ext_vector_type(2)));
typedef unsigned int u32x4 __attribute__((ext_vector_type(4)));
typedef int  i32x8 __attribute__((ext_vector_type(8)));
typedef int  i32x4 __attribute__((ext_vector_type(4)));

// ---- wave32 sum using the matrix unit: D = A x 1s  =>  row sums ------------
// A (16x4 f32): lane L (0..15) holds {K=0,K=1}, lane L+16 holds {K=2,K=3} for
// row M=L. Putting v in .x and 0 in .y gives D[m][*] = v_m + v_{m+16}.
// D layout: lane n(0..15) VGPR r = D[M=r][n]; lanes 16..31 VGPR r = D[M=r+8][n].
// Sum of the 8 D regs + xor-16 shuffle = full 32-lane sum (all lanes).
__device__ __forceinline__ float wave_sum(float v) {
#if defined(__gfx1250__) && __has_builtin(__builtin_amdgcn_wmma_f32_16x16x4_f32)
    v2f a; a.x = v;   a.y = 0.0f;
    v2f b; b.x = 1.0f; b.y = 1.0f;
    v8f c = {};
    v8f d = __builtin_amdgcn_wmma_f32_16x16x4_f32(
        /*neg_a=*/false, a, /*neg_b=*/false, b,
        /*c_mod=*/(short)0, c, /*reuse_a=*/false, /*reuse_b=*/false);
    float s = d[0] + d[1] + d[2] + d[3] + d[4] + d[5] + d[6] + d[7];
    s += __shfl_xor(s, 16, 32);
    return s;
#else
    for (int off = 16; off > 0; off >>= 1) v += __shfl_xor(v, off, 32);
    return v;
#endif
}

// ---- Kernel 1: best_truth_overlap[p] = max_t IoU(t,p); also zero flags -----
__global__ __launch_bounds__(256) void k_bto(
    const float* __restrict__ locs, const float* __restrict__ params,
    const float* __restrict__ truths, float* __restrict__ bto,
    unsigned char* __restrict__ flags, int P, int T)
{
    __shared__ float sbox[MAX_T * 4];
    __shared__ float sarea[MAX_T];
    const int tid = threadIdx.x;

#if defined(__gfx1250__) && __has_builtin(__builtin_amdgcn_tensor_load_to_lds)
    if (tid < 32) {  // one wave issues the TDM copy: truths -> LDS (T*16 bytes)
        uint64_t gaddr = (uint64_t)(uintptr_t)truths;
        uint32_t laddr = (uint32_t)(uintptr_t)(&sbox[0]);  // low 32b = LDS offset
        uint32_t nelem = (uint32_t)(T * 4);                // f32 elements
        u32x4 g0;
        g0[0] = 1u;                                        // count=1, user mode
        g0[1] = laddr;                                     // lds_addr [63:32]
        g0[2] = (uint32_t)(gaddr & 0xFFFFFFFFu);           // global_addr [95:64]
        g0[3] = (uint32_t)((gaddr >> 32) & 0x01FFFFFFu)    // global_addr [120:96]
              | (2u << 30);                                // type=2 ("image")
        i32x8 g1;
        g1[0] = (int)(2u << 16);                           // mask=0, data_size=4B
        g1[1] = (int)((nelem & 0xFFFFu) << 16);            // tensor_dim0[15:0]
        g1[2] = (int)((nelem >> 16) | (1u << 16));         // tensor_dim0 hi, tensor_dim1=1
        g1[3] = (int)((nelem & 0xFFFFu) << 16);            // tile_dim0 = nelem
        g1[4] = 1;                                         // tile_dim1=1, tile_dim2=0
        g1[5] = (int)nelem;                                // tensor_dim0_stride lo
        g1[6] = 0;
        g1[7] = 0;
        i32x4 gz4  = {0, 0, 0, 0};
        i32x8 gz8  = {0, 0, 0, 0, 0, 0, 0, 0};
        __builtin_amdgcn_tensor_load_to_lds(g0, g1, gz4, gz4, gz8, 0);
        __builtin_amdgcn_s_wait_tensorcnt(0);
    }
#else
    for (int i = tid; i < T * 4; i += blockDim.x) sbox[i] = truths[i];
#endif
    __syncthreads();
    for (int t = tid; t < T; t += blockDim.x)
        sarea[t] = (sbox[4*t+2] - sbox[4*t+0]) * (sbox[4*t+3] - sbox[4*t+1]);
    __syncthreads();

    int p = blockIdx.x * blockDim.x + tid;
    if (p >= P) return;

    float cx = locs[2*p], cy = locs[2*p+1];
    float hw = 0.5f * params[3*p], hh = 0.5f * params[3*p+1];
    float px1 = cx - hw, py1 = cy - hh, px2 = cx + hw, py2 = cy + hh;
    float pa  = (px2 - px1) * (py2 - py1);

    // track max IoU as (inter, denom) pair: a/b > c/d  <=>  a*d > c*b  (all > 0)
    float bi = 0.0f, bd = 1.0f;
    for (int t = 0; t < T; ++t) {
        float x1 = sbox[4*t+0], y1 = sbox[4*t+1];
        float x2 = sbox[4*t+2], y2 = sbox[4*t+3];
        float iw = fmaxf(fminf(x2, px2) - fmaxf(x1, px1), 0.0f);
        float ih = fmaxf(fminf(y2, py2) - fmaxf(y1, py1), 0.0f);
        float inter = iw * ih;
        float den   = sarea[t] + pa - inter;
        bool  g = inter * bd > bi * den;
        bi = g ? inter : bi;
        bd = g ? den   : bd;
    }
    bto[p]   = bi / bd;
    flags[p] = 0;
}

// ---- Kernel 2: per-truth max + argmax over all priors ----------------------
__global__ __launch_bounds__(256) void k_bpi(
    const float* __restrict__ locs, const float* __restrict__ params,
    const float* __restrict__ truths, float* __restrict__ bpo,
    int* __restrict__ bpi, int P)
{
    const int t   = blockIdx.x;
    const int tid = threadIdx.x;
    float x1 = truths[4*t+0], y1 = truths[4*t+1];
    float x2 = truths[4*t+2], y2 = truths[4*t+3];
    float ta = (x2 - x1) * (y2 - y1);

    float bi = 0.0f, bd = 1.0f; int bx = 0;
    for (int p = tid; p < P; p += blockDim.x) {
        float cx = locs[2*p], cy = locs[2*p+1];
        float hw = 0.5f * params[3*p], hh = 0.5f * params[3*p+1];
        float px1 = cx - hw, py1 = cy - hh, px2 = cx + hw, py2 = cy + hh;
        float pa  = (px2 - px1) * (py2 - py1);
        float iw = fmaxf(fminf(x2, px2) - fmaxf(x1, px1), 0.0f);
        float ih = fmaxf(fminf(y2, py2) - fmaxf(y1, py1), 0.0f);
        float inter = iw * ih;
        float den   = ta + pa - inter;
        bool  g = inter * bd > bi * den;     // strict ">" keeps lowest index
        if (g) { bi = inter; bd = den; bx = p; }
    }

    __shared__ float si[256], sd[256];
    __shared__ int   sx[256];
    si[tid] = bi; sd[tid] = bd; sx[tid] = bx;
    __syncthreads();
    for (int s = 128; s > 0; s >>= 1) {
        if (tid < s) {
            float oi = si[tid+s], od = sd[tid+s]; int ox = sx[tid+s];
            float l = oi * sd[tid], r = si[tid] * od;
            bool  g = (l > r) || (l == r && ox < sx[tid]);  // tie -> lower idx
            if (g) { si[tid] = oi; sd[tid] = od; sx[tid] = ox; }
        }
        __syncthreads();
    }
    if (tid == 0) { bpo[t] = si[0] / sd[0]; bpi[t] = sx[0]; }
}

// ---- Kernel 3: sequential scatter (last-t-wins) + final reduction ----------
__global__ __launch_bounds__(256) void k_final(
    const float* __restrict__ params, float* __restrict__ bto,
    unsigned char* __restrict__ flags, const float* __restrict__ bpo,
    const int* __restrict__ bpi, float* __restrict__ out, int P, int T)
{
    const int tid = threadIdx.x;
    if (tid == 0) {
        for (int t = 0; t < T; ++t) {      // deterministic order: last t wins
            int p = bpi[t];
            bto[p]   = bpo[t];
            flags[p] = 1;
        }
    }
    __threadfence();
    __syncthreads();

    float num = 0.0f, den = 0.0f;
    for (int p = tid; p < P; p += blockDim.x) {
        float a = params[3*p + 2];
        float s = 1.0f / (1.0f + __expf(-a));     // sigmoid
        float b = bto[p];
        float f = flags[p] ? 5.0f : (b > 0.5f ? 1.0f : 0.0f);  // K / THRESH
        num = fmaf(s * f, b, num) + s;            // + BETA * s  (BETA = 1)
        den += f;
    }
    float wn = wave_sum(num);
    float wd = wave_sum(den);

    __shared__ float swn[8], swd[8];
    if ((tid & 31) == 0) { swn[tid >> 5] = wn; swd[tid >> 5] = wd; }
    __syncthreads();
    if (tid == 0) {
        float N = 0.0f, D = 0.0f;
        int nw = (int)(blockDim.x + 31) >> 5;
        for (int w = 0; w < nw; ++w) { N += swn[w]; D += swd[w]; }
        out[0] = N / D;
    }
}

extern "C" void kernel_launch(void* const* d_in, const int* in_sizes, int n_in,
                              void* d_out, int out_size, void* d_ws, size_t ws_size,
                              hipStream_t stream) {
    const float* locs   = (const float*)d_in[0];  // [P,2]
    const float* params = (const float*)d_in[1];  // [P,3]
    const float* truths = (const float*)d_in[2];  // [T,4]
    int P = in_sizes[0] / 2;
    int T = in_sizes[2] / 4;
    if (T > MAX_T) T = MAX_T;

    char* ws = (char*)d_ws;
    float*         bto   = (float*)ws;                                  // P f32
    float*         bpo   = (float*)(ws + (size_t)P * 4);                // T f32
    int*           bpi   = (int*)  (ws + (size_t)P * 4 + (size_t)T * 4);// T i32
    unsigned char* flags = (unsigned char*)(ws + (size_t)P * 4 + (size_t)T * 8);

    dim3 blk(256);
    k_bto  <<<dim3((P + 255) / 256), blk, 0, stream>>>(locs, params, truths, bto, flags, P, T);
    k_bpi  <<<dim3(T),               blk, 0, stream>>>(locs, params, truths, bpo, bpi, P);
    k_final<<<dim3(1),               blk, 0, stream>>>(params, bto, flags, bpo, bpi,
                                                       (float*)d_out, P, T);
}